// CompGCNConvDGL_9062380994847
// MI455X (gfx1250) — compile-verified
//
#include <hip/hip_runtime.h>
#include <hip/hip_bf16.h>
#include <stddef.h>

typedef __attribute__((ext_vector_type(2))) float v2f;
typedef __attribute__((ext_vector_type(4))) float v4f;
typedef __attribute__((ext_vector_type(8))) float v8f;

#define D 64            // D_IN == D_OUT == 64
#define KSTEPS 16       // 64 / 4 (K per V_WMMA_F32_16X16X4_F32)

// ---------------------------------------------------------------------------
// Kernel 1: rel_out = rel_emb @ W_rel^T  (-> d_out tail)
//           re_t    = rel_emb @ W_fwd^T  (-> ws)
// Tiny (200x64 rows); plain VALU dot products, one output element per thread.
// ---------------------------------------------------------------------------
__global__ __launch_bounds__(256)
void rel_tables_kernel(const float* __restrict__ rel_emb,
                       const float* __restrict__ W_rel,
                       const float* __restrict__ W_fwd,
                       float* __restrict__ rel_out,
                       float* __restrict__ re_t,
                       int n_rel) {
  int idx = blockIdx.x * 256 + threadIdx.x;
  if (idx >= n_rel * D) return;
  int r = idx >> 6;
  int o = idx & 63;
  const float* e  = rel_emb + (size_t)r * D;
  const float* wr = W_rel   + (size_t)o * D;
  const float* wf = W_fwd   + (size_t)o * D;
  float accr = 0.f, accf = 0.f;
#pragma unroll 8
  for (int k = 0; k < D; ++k) {
    float x = e[k];
    accr += x * wr[k];
    accf += x * wf[k];
  }
  rel_out[idx] = accr;
  re_t[idx]    = accf;
}

// ---------------------------------------------------------------------------
// Kernel 2: fused node GEMMs via V_WMMA_F32_16X16X4_F32.
//   nf_t     = node_feat @ W_fwd^T            (-> ws, edge-phase table)
//   out_self = node_feat @ W_self^T + bias    (-> d_out, scatter base)
// One wave per 16-row strip. 4 N-tiles (64 cols) x 2 weights = 8 accumulators.
// B tiles (W^T) staged in LDS as (k,k+1) float2 pairs: Wp[kp][o].
// ---------------------------------------------------------------------------
__global__ __launch_bounds__(256)
void node_gemm_wmma_kernel(const float* __restrict__ node_feat,
                           const float* __restrict__ W_self,
                           const float* __restrict__ W_fwd,
                           const float* __restrict__ bias,
                           float* __restrict__ out_self,
                           float* __restrict__ nf_t,
                           int n_rows) {
  __shared__ v2f Wp_f[32 * 64];   // 16 KB: W_fwd^T  k-pairs, [kp][o]
  __shared__ v2f Wp_s[32 * 64];   // 16 KB: W_self^T k-pairs, [kp][o]

  const int t = threadIdx.x;
  // Stage both weight matrices (transposed, pair-packed along K) into LDS.
  for (int i = t; i < 32 * 64; i += 256) {
    int kp = i >> 6;        // K pair index: covers k = 2kp, 2kp+1
    int o  = i & 63;        // output column
    v2f wf, ws;
    wf.x = W_fwd [(size_t)o * D + 2 * kp];
    wf.y = W_fwd [(size_t)o * D + 2 * kp + 1];
    ws.x = W_self[(size_t)o * D + 2 * kp];
    ws.y = W_self[(size_t)o * D + 2 * kp + 1];
    Wp_f[i] = wf;
    Wp_s[i] = ws;
  }
  __syncthreads();

  const int wave  = t >> 5;                 // wave32
  const int lane  = t & 31;
  const int strip = blockIdx.x * 8 + wave;  // 16 rows per strip
  if (strip * 16 >= n_rows) return;         // wave-uniform: EXEC stays all-1s

  const int m0    = strip * 16;
  const int m     = lane & 15;              // A-matrix M = lane % 16
  const int khalf = lane >> 4;              // A-matrix K pair select (ISA 16x4 f32 layout)

  const float* arow = node_feat + (size_t)(m0 + m) * D;

  v8f acc_f[4] = {};   // W_fwd  accumulators, N-tiles 0..3
  v8f acc_s[4] = {};   // W_self accumulators

  for (int s = 0; s < KSTEPS; ++s) {
    const int k0 = 4 * s + 2 * khalf;       // this lane's K pair within step
    const int kp = 2 * s + khalf;
    v2f a = *(const v2f*)(arow + k0);       // A[m, k0], A[m, k0+1]
#pragma unroll
    for (int n = 0; n < 4; ++n) {
      // B[k, o] = W^T: lanes 0-15 -> K=4s..,N=16n+lane; lanes 16-31 -> K=4s+2..
      v2f bf = Wp_f[kp * 64 + n * 16 + m];
      v2f bs = Wp_s[kp * 64 + n * 16 + m];
      acc_f[n] = __builtin_amdgcn_wmma_f32_16x16x4_f32(
          false, a, false, bf, (short)0, acc_f[n], false, false);
      acc_s[n] = __builtin_amdgcn_wmma_f32_16x16x4_f32(
          false, a, false, bs, (short)0, acc_s[n], false, false);
    }
  }

  // C/D layout: VGPR v -> row m0 + v + 8*khalf, col = 16n + (lane&15).
#pragma unroll
  for (int n = 0; n < 4; ++n) {
    const int col = n * 16 + m;
    const float b = bias[col];
#pragma unroll
    for (int v = 0; v < 8; ++v) {
      const int row = m0 + v + khalf * 8;
      nf_t    [(size_t)row * D + col] = acc_f[n][v];
      out_self[(size_t)row * D + col] = acc_s[n][v] + b;
    }
  }
}

// ---------------------------------------------------------------------------
// Kernel 3: edge scatter. 16 lanes per edge, float4 per lane.
//   if direction==0: out[dst] += nf_t[src] - re_t[etype]   (hardware f32 atomics)
// All tables are L2-resident (nf_t 25.6MB, out 25.6MB << 192MB L2).
// ---------------------------------------------------------------------------
__global__ __launch_bounds__(256)
void edge_scatter_kernel(const int* __restrict__ src,
                         const int* __restrict__ dst,
                         const int* __restrict__ etype,
                         const int* __restrict__ dir,
                         const float* __restrict__ nf_t,
                         const float* __restrict__ re_t,
                         float* __restrict__ out,
                         int n_edges) {
  const int tid  = blockIdx.x * 256 + threadIdx.x;
  const int e    = tid >> 4;
  const int part = tid & 15;
  if (e >= n_edges) return;
  if (dir[e] != 0) return;                  // only forward edges contribute

  const int s = src[e];
  const int d = dst[e];
  const int r = etype[e];

  v4f a = *(const v4f*)(nf_t + (size_t)s * D + part * 4);
  v4f b = *(const v4f*)(re_t + (size_t)r * D + part * 4);
  v4f msg = a - b;

  float* p = out + (size_t)d * D + part * 4;
#pragma unroll
  for (int i = 0; i < 4; ++i) {
    // Guaranteed hardware f32 atomic (non-returning -> STOREcnt), no CAS loop.
    asm volatile("global_atomic_add_f32 %0, %1, off"
                 :: "v"(p + i), "v"(msg[i])
                 : "memory");
  }
}

// ---------------------------------------------------------------------------
// Launcher. Inputs (setup_inputs order):
//  0 node_feat [N,64]  1 rel_emb [R,64]  2 src [E]  3 dst [E]  4 etype [E]
//  5 direction [E]     6 W_self [64,64]  7 W_forward  8 W_backward
//  9 W_rel            10 bias [64]
// d_out: [N*64 out | R*64 rel_out].  ws: [R*64 re_t | N*64 nf_t] (floats).
// ---------------------------------------------------------------------------
extern "C" void kernel_launch(void* const* d_in, const int* in_sizes, int n_in,
                              void* d_out, int out_size, void* d_ws, size_t ws_size,
                              hipStream_t stream) {
  const float* node_feat = (const float*)d_in[0];
  const float* rel_emb   = (const float*)d_in[1];
  const int*   src       = (const int*)d_in[2];
  const int*   dst       = (const int*)d_in[3];
  const int*   etype     = (const int*)d_in[4];
  const int*   dir       = (const int*)d_in[5];
  const float* W_self    = (const float*)d_in[6];
  const float* W_fwd     = (const float*)d_in[7];
  const float* W_rel     = (const float*)d_in[9];
  const float* bias      = (const float*)d_in[10];

  const int n_nodes = in_sizes[0] / D;
  const int n_rel   = in_sizes[1] / D;
  const int n_edges = in_sizes[2];

  float* out     = (float*)d_out;                       // [n_nodes, 64]
  float* rel_out = out + (size_t)n_nodes * D;           // [n_rel, 64]
  float* re_t    = (float*)d_ws;                        // [n_rel, 64]
  float* nf_t    = re_t + (size_t)n_rel * D;            // [n_nodes, 64]

  // 1) rel tables (rel_out to d_out, re_t to ws)
  {
    int total = n_rel * D;
    rel_tables_kernel<<<(total + 255) / 256, 256, 0, stream>>>(
        rel_emb, W_rel, W_fwd, rel_out, re_t, n_rel);
  }

  // 2) fused node GEMMs (WMMA f32): nf_t to ws, self+bias base into d_out
  {
    int strips = (n_nodes + 15) / 16;
    int blocks = (strips + 7) / 8;
    node_gemm_wmma_kernel<<<blocks, 256, 0, stream>>>(
        node_feat, W_self, W_fwd, bias, out, nf_t, n_nodes);
  }

  // 3) edge scatter with hardware f32 atomics into d_out
  {
    long long threads = (long long)n_edges * 16;
    int blocks = (int)((threads + 255) / 256);
    edge_scatter_kernel<<<blocks, 256, 0, stream>>>(
        src, dst, etype, dir, nf_t, re_t, out, n_edges);
  }
}